// Custom_Loss2_83322365542989
// MI455X (gfx1250) — compile-verified
//
#include <hip/hip_runtime.h>
#include <hip/hip_bf16.h>
#include <stdint.h>

// ---------------------------------------------------------------------------
// Weighted-CE + MSE fused loss, MI455X (gfx1250, wave32).
// Memory-bound: ~188 MB read once -> ~8 us floor @ 23.3 TB/s.
// CDNA5 paths:
//   * GLOBAL_LOAD_ASYNC_TO_LDS_B128 (ASYNCcnt) streams the [B,9] logits
//     straight into LDS, no VGPR round-trip; s_wait_asynccnt + barrier.
//   * V_WMMA_F32_16X16X4_F32 as a full-precision 32-lane reduction engine
//     (A = lane partials in K0/K2, B = ones, C = 0).
// ---------------------------------------------------------------------------

typedef __attribute__((ext_vector_type(2))) float v2f;
typedef __attribute__((ext_vector_type(8))) float v8f;

#define TPB        256
#define WAVES      (TPB / 32)
#define NC         9
#define ROWS_TILE  1024                       // rows staged per block tile
#define F4_TILE    (ROWS_TILE * NC / 4)       // 2304 float4 chunks per tile
#define CHUNKS_PT  (F4_TILE / TPB)            // 9 b128 async copies / thread
#define ROWS_PT    (ROWS_TILE / TPB)          // 4 rows computed / thread
#define NBLOCKS    2048

// Wave32 all-lane sum via V_WMMA_F32_16X16X4_F32 (full f32 precision).
// A[m][0]=x(lane m), A[m][2]=x(lane m+16), other K zero; B = ones =>
// D[m][n] = x[m] + x[m+16].  Per-lane sum of the 8 D VGPRs gives the
// row-half totals; xor-16 shuffle combines them.  EXEC all-1s at call site.
__device__ __forceinline__ float wave_sum(float x) {
#if defined(__has_builtin) && __has_builtin(__builtin_amdgcn_wmma_f32_16x16x4_f32)
  v2f a;  a.x = x;    a.y = 0.0f;
  v2f b;  b.x = 1.0f; b.y = 1.0f;
  v8f c = {};
  v8f d = __builtin_amdgcn_wmma_f32_16x16x4_f32(
      /*neg_a=*/false, a, /*neg_b=*/false, b,
      /*c_mod=*/(short)0, c, /*reuse_a=*/false, /*reuse_b=*/false);
  float s = d[0] + d[1] + d[2] + d[3] + d[4] + d[5] + d[6] + d[7];
  s += __shfl_xor(s, 16, 32);
  return s;
#else
  for (int off = 16; off > 0; off >>= 1) x += __shfl_xor(x, off, 32);
  return x;
#endif
}

// Per-row weighted cross entropy from a 9-float logit row.
__device__ __forceinline__ float row_wce(const float* __restrict__ x, int t) {
  float m = x[0];
  int   am = 0;
#pragma unroll
  for (int c = 1; c < NC; ++c) {            // first-max == jnp.argmax
    const float v = x[c];
    if (v > m) { m = v; am = c; }
  }
  float se = 0.0f;
#pragma unroll
  for (int c = 0; c < NC; ++c) se += __expf(x[c] - m);
  const float ce_i = (m + __logf(se)) - x[t];   // -log_softmax[t]
  const float w    = fabsf((float)(am - t)) + 1.0f;
  return w * ce_i;
}

__global__ void loss_zero_kernel(float* __restrict__ acc) {
  if (threadIdx.x < 2) acc[threadIdx.x] = 0.0f;
}

__global__ __launch_bounds__(TPB) void loss_reduce_kernel(
    const float* __restrict__ probs,   // [B, 9]
    const int*   __restrict__ tclass,  // [B]
    const float* __restrict__ pp,      // [B]
    const float* __restrict__ tp,      // [B]
    float*       __restrict__ acc,     // acc[0]=sum(w*ce), acc[1]=sum(dp^2)
    int B) {
  alignas(16) __shared__ float smem[ROWS_TILE * NC];   // 36 KB of 320 KB LDS
  __shared__ float swce[WAVES];
  __shared__ float ssq[WAVES];

  const int tid = threadIdx.x;
  float acc_wce = 0.0f;
  float acc_sq  = 0.0f;

  // Low 32 bits of a generic shared pointer == LDS byte address (aperture
  // mapping keeps the LDS offset in the low bits); this is what the async-copy VDST wants.
  const uint32_t ldsBase = (uint32_t)(uintptr_t)&smem[0];

  // ---- Phase 1a: full 1024-row tiles, async global->LDS b128 staging ----
  const int nFull = B / ROWS_TILE;
  for (int tile = blockIdx.x; tile < nFull; tile += gridDim.x) {
    const float* gbase = probs + (size_t)tile * (ROWS_TILE * NC);
#pragma unroll
    for (int k = 0; k < CHUNKS_PT; ++k) {
      const uint32_t voff = (uint32_t)((k * TPB + tid) * 16);  // byte offset
      const uint32_t laddr = ldsBase + voff;
      // GVS mode: mem = SGPR64(gbase) + VGPR32(voff); LDS dst = VGPR(laddr).
      asm volatile("global_load_async_to_lds_b128 %0, %1, %2 offset:0"
                   :
                   : "v"(laddr), "v"(voff), "s"(gbase)
                   : "memory");
    }
    asm volatile("s_wait_asynccnt 0x0" ::: "memory");  // this wave's copies
    __syncthreads();                                   // everyone's copies

#pragma unroll
    for (int j = 0; j < ROWS_PT; ++j) {
      const int rloc = j * TPB + tid;                  // row within tile
      const int row  = tile * ROWS_TILE + rloc;
      acc_wce += row_wce(&smem[rloc * NC], tclass[row]);
    }
    __syncthreads();                                   // before next overwrite
  }

  // ---- Phase 1b: ragged tail rows (< 1024), direct from global ----
  for (int row = nFull * ROWS_TILE + blockIdx.x * TPB + tid; row < B;
       row += gridDim.x * TPB) {
    float x[NC];
#pragma unroll
    for (int c = 0; c < NC; ++c) x[c] = probs[row * NC + c];
    acc_wce += row_wce(x, tclass[row]);
  }

  // ---- Phase 2: MSE partials, b128 loads + scalar tail ----
  const int n4 = B >> 2;
  const float4* __restrict__ pp4 = (const float4*)pp;
  const float4* __restrict__ tp4 = (const float4*)tp;
  for (int i = blockIdx.x * TPB + tid; i < n4; i += gridDim.x * TPB) {
    const float4 a = pp4[i];
    const float4 b = tp4[i];
    float d;
    d = a.x - b.x; acc_sq += d * d;
    d = a.y - b.y; acc_sq += d * d;
    d = a.z - b.z; acc_sq += d * d;
    d = a.w - b.w; acc_sq += d * d;
  }
  for (int i = (n4 << 2) + blockIdx.x * TPB + tid; i < B;
       i += gridDim.x * TPB) {
    const float d = pp[i] - tp[i];
    acc_sq += d * d;
  }

  // ---- Hierarchical reduction: WMMA intra-wave, LDS cross-wave ----
  const float wce = wave_sum(acc_wce);
  const float sq  = wave_sum(acc_sq);
  const int wave = tid >> 5;
  const int lane = tid & 31;
  if (lane == 0) { swce[wave] = wce; ssq[wave] = sq; }
  __syncthreads();
  if (tid == 0) {
    float a = 0.0f, b = 0.0f;
#pragma unroll
    for (int w = 0; w < WAVES; ++w) { a += swce[w]; b += ssq[w]; }
    atomicAdd(&acc[0], a);
    atomicAdd(&acc[1], b);
  }
}

__global__ void loss_finalize_kernel(const float* __restrict__ acc,
                                     float* __restrict__ out, float invB) {
  if (threadIdx.x == 0) {
    const float ce  = acc[0] * invB;   // sum(w*ce)/B
    const float mse = acc[1] * invB;
    out[0] = mse + ce;                 // ALPHA = 1.0
    out[1] = mse;
    out[2] = ce;
  }
}

extern "C" void kernel_launch(void* const* d_in, const int* in_sizes, int n_in,
                              void* d_out, int out_size, void* d_ws, size_t ws_size,
                              hipStream_t stream) {
  const float* probs  = (const float*)d_in[0];  // pred_probs [B,9] f32
  const int*   tclass = (const int*)  d_in[1];  // true_class [B]
  const float* pp     = (const float*)d_in[2];  // pred_points [B,1] f32
  const float* tp     = (const float*)d_in[3];  // true_points [B,1] f32
  const int B = in_sizes[1];                    // true_class element count
  float* acc = (float*)d_ws;                    // 2 f32 accumulators
  float* out = (float*)d_out;                   // (loss, mse, ce)

  loss_zero_kernel<<<1, 32, 0, stream>>>(acc);
  int blocks = (B + ROWS_TILE - 1) / ROWS_TILE;
  if (blocks < 1) blocks = 1;
  if (blocks > NBLOCKS) blocks = NBLOCKS;
  loss_reduce_kernel<<<blocks, TPB, 0, stream>>>(probs, tclass, pp, tp, acc, B);
  loss_finalize_kernel<<<1, 32, 0, stream>>>(acc, out, 1.0f / (float)B);
}